// GraphConv_85650237816948
// MI455X (gfx1250) — compile-verified
//
#include <hip/hip_runtime.h>
#include <hip/hip_bf16.h>

// ---------------------------------------------------------------------------
// GraphConv for MI455X (gfx1250, wave32):
//   x1 = scatter_add over edges of v * x[c]      (SpMM, L2 atomic phase)
//   out = x1 @ W.T + b                           (GEMM, V_WMMA_F32_16X16X4_F32,
//                                                 async global->LDS staging)
// ---------------------------------------------------------------------------

#define D 128            // D_IN == D_OUT == 128
#define LDS_PAD 132      // 128 + 4 pad -> bank stride 4/row, conflict-free b64 reads

typedef __attribute__((ext_vector_type(2))) float v2f;
typedef __attribute__((ext_vector_type(8))) float v8f;

// ---------------- Kernel 1: zero the x1 accumulator in workspace ----------
__global__ __launch_bounds__(256) void gc_zero_ws(float4* __restrict__ ws, long n4) {
    long i = (long)blockIdx.x * blockDim.x + threadIdx.x;
    long stride = (long)gridDim.x * blockDim.x;
    float4 z = make_float4(0.f, 0.f, 0.f, 0.f);
    for (; i < n4; i += stride) ws[i] = z;
}

// ---------------- Kernel 2: edge-parallel SpMM with fp32 atomics ----------
// One wave (32 lanes) per edge; lane l handles dims [4l, 4l+4).
// Edge descriptors are made wave-uniform via readfirstlane so they lower to
// s_load_b32 (scalar cache / KMcnt) instead of 32 redundant vector loads.
// x[c] gather is a coalesced 512B/wave global_load_b128; x (51MB) and
// x1 (51MB) are L2-resident (192MB L2), so atomics resolve in L2 atomic units.
__global__ __launch_bounds__(256) void gc_spmm(const float* __restrict__ x,
                                               const int* __restrict__ erow,
                                               const int* __restrict__ ecol,
                                               const float* __restrict__ eval,
                                               float* __restrict__ x1,
                                               int nEdges) {
    int wave = (blockIdx.x * blockDim.x + threadIdx.x) >> 5;
    if (wave >= nEdges) return;
    int e = __builtin_amdgcn_readfirstlane(wave);   // provably wave-uniform
    int lane = threadIdx.x & 31;

    int r = erow[e];                                // s_load_b32
    int c = ecol[e];                                // s_load_b32
    float v = eval[e];                              // s_load_b32

    const float4* src = (const float4*)(x + (long)c * D);
    float4 g = src[lane];                           // global_load_b128
    float* dst = x1 + (long)r * D + lane * 4;
#if defined(__AMDGCN__)
    unsafeAtomicAdd(dst + 0, g.x * v);              // global_atomic_add_f32
    unsafeAtomicAdd(dst + 1, g.y * v);
    unsafeAtomicAdd(dst + 2, g.z * v);
    unsafeAtomicAdd(dst + 3, g.w * v);
#else
    atomicAdd(dst + 0, g.x * v);
    atomicAdd(dst + 1, g.y * v);
    atomicAdd(dst + 2, g.z * v);
    atomicAdd(dst + 3, g.w * v);
#endif
}

// ---------------- Kernel 3: out = x1 @ W.T + b via WMMA f32 16x16x4 -------
// Block = 256 threads = 8 waves. Block b owns rows [16b, 16b+16).
// Wave w owns output columns [16w, 16w+16). K swept 4 at a time, 32 WMMAs.
// A tile (8KB) and W (64KB) are staged into padded LDS with CDNA5 async
// global->LDS copies (ASYNCcnt), so the shared A tile is read from HBM/L2
// once per block instead of once per wave.
__global__ __launch_bounds__(256) void gc_gemm_wmma(const float* __restrict__ x1,
                                                    const float* __restrict__ W,
                                                    const float* __restrict__ bias,
                                                    float* __restrict__ out) {
    __shared__ __attribute__((aligned(16))) float ldsW[D * LDS_PAD];   // 66 KB
    __shared__ __attribute__((aligned(16))) float ldsA[16 * LDS_PAD];  //  8.25 KB

    int row0 = blockIdx.x * 16;                // 6250 * 16 == 100000 exactly

    unsigned long long aBase = (unsigned long long)(uintptr_t)(x1 + (long)row0 * D);
    unsigned long long wBase = (unsigned long long)(uintptr_t)W;

    // ---- Async stage A tile: 16 rows x 128 f32 = 512 x 16B chunks --------
    #pragma unroll
    for (int rep = 0; rep < 2; ++rep) {
        int i = threadIdx.x + rep * 256;       // 0..511
        int row = i >> 5, chunk = i & 31;      // chunk = 16B of the row
        unsigned long long gsrc = aBase + (unsigned long long)((row * D + chunk * 4) * 4);
        unsigned ldst = (unsigned)(uintptr_t)&ldsA[row * LDS_PAD + chunk * 4];
        asm volatile("global_load_async_to_lds_b128 %0, %1, off"
                     :: "v"(ldst), "v"(gsrc) : "memory");
    }
    // ---- Async stage W: 128 rows x 128 f32 = 4096 x 16B chunks -----------
    #pragma unroll
    for (int rep = 0; rep < 16; ++rep) {
        int i = threadIdx.x + rep * 256;       // 0..4095
        int row = i >> 5, chunk = i & 31;
        unsigned long long gsrc = wBase + (unsigned long long)((row * D + chunk * 4) * 4);
        unsigned ldst = (unsigned)(uintptr_t)&ldsW[row * LDS_PAD + chunk * 4];
        asm volatile("global_load_async_to_lds_b128 %0, %1, off"
                     :: "v"(ldst), "v"(gsrc) : "memory");
    }
    asm volatile("s_wait_asynccnt 0" ::: "memory");  // all async copies in LDS
    __syncthreads();

    int lane   = threadIdx.x & 31;
    int waveId = threadIdx.x >> 5;
    int n0     = waveId * 16;

    int halfSel = lane >> 4;                   // 0: lanes 0-15, 1: lanes 16-31
    int m16     = lane & 15;
    int ncol    = n0 + m16;                    // B/C/D: N = lane%16
    int kgrp    = halfSel * 2;                 // A/B: lanes 16-31 carry K+2,K+3

    const float* aPtr = ldsA + m16 * LDS_PAD + kgrp;    // A: M = lane%16
    const float* bPtr = ldsW + ncol * LDS_PAD + kgrp;

    v8f acc = {};
    #pragma unroll
    for (int k0 = 0; k0 < D; k0 += 4) {
        v2f a  = *(const v2f*)(aPtr + k0);     // ds_load_b64, conflict-free
        v2f bm = *(const v2f*)(bPtr + k0);     // ds_load_b64, conflict-free
        // D = A(16x4) x B(4x16) + C ; full fp32, matches reference precision
        acc = __builtin_amdgcn_wmma_f32_16x16x4_f32(
            /*neg_a=*/false, a, /*neg_b=*/false, bm,
            /*c_mod=*/(short)0, acc, /*reuse_a=*/false, /*reuse_b=*/false);
    }

    // C/D layout: VGPR i -> M = row0 + 8*halfSel + i, N = ncol.
    float bv = bias[ncol];
    int mbase = row0 + halfSel * 8;
    #pragma unroll
    for (int i = 0; i < 8; ++i) {
        out[(long)(mbase + i) * D + ncol] = acc[i] + bv;
    }
}

// ---------------------------------------------------------------------------
extern "C" void kernel_launch(void* const* d_in, const int* in_sizes, int n_in,
                              void* d_out, int out_size, void* d_ws, size_t ws_size,
                              hipStream_t stream) {
    const float* x    = (const float*)d_in[0];
    const int*   erow = (const int*)  d_in[1];
    const int*   ecol = (const int*)  d_in[2];
    const float* eval = (const float*)d_in[3];
    const float* W    = (const float*)d_in[4];
    const float* bias = (const float*)d_in[5];
    float* out = (float*)d_out;

    int nNodes = in_sizes[0] / D;              // 100000
    int nEdges = in_sizes[1];                  // 3200000

    float* x1 = (float*)d_ws;                  // N*D floats = 51.2 MB scratch

    // 1) zero x1
    long n4 = (long)nNodes * D / 4;
    int zblocks = (int)((n4 + 255) / 256);
    if (zblocks > 4096) zblocks = 4096;        // grid-stride
    gc_zero_ws<<<zblocks, 256, 0, stream>>>((float4*)x1, n4);

    // 2) SpMM: one wave per edge, 8 edges per 256-thread block
    int sblocks = (nEdges + 7) / 8;
    gc_spmm<<<sblocks, 256, 0, stream>>>(x, erow, ecol, eval, x1, nEdges);

    // 3) GEMM: 16 rows per block, 8 column-tile waves per block
    int gblocks = nNodes / 16;                 // 6250
    gc_gemm_wmma<<<gblocks, 256, 0, stream>>>(x1, W, bias, out);
}